// MPALayer_71743133712744
// MI455X (gfx1250) — compile-verified
//
#include <hip/hip_runtime.h>
#include <hip/hip_bf16.h>

// ---------------- problem constants (from reference) ----------------
#define NN   50000
#define EE   400000
#define PP   3
#define DIN  128
#define HH   8
#define OO   32
#define HO   256      // H*O
#define HID  128
#define NEG_SLOPE 0.2f

typedef __attribute__((ext_vector_type(2))) float v2f;
typedef __attribute__((ext_vector_type(8))) float v8f;

// f32 WMMA 16x16x4: D = A(16x4) * B(4x16) + C(16x16), all f32, exact-precision path.
#define WMMA_F32(a, b, c) \
  __builtin_amdgcn_wmma_f32_16x16x4_f32(false, (a), false, (b), (short)0, (c), false, false)

// ---- order-preserving f32 <-> u32 encoding for exact atomic segment-max ----
__device__ __forceinline__ unsigned enc_f32(float x) {
  unsigned b = __float_as_uint(x);
  return (b & 0x80000000u) ? ~b : (b | 0x80000000u);
}
__device__ __forceinline__ float dec_f32(unsigned e) {
  unsigned b = (e & 0x80000000u) ? (e & 0x7FFFFFFFu) : ~e;
  return __uint_as_float(b);
}
__device__ __forceinline__ float leaky(float x) {
  return x >= 0.f ? x : NEG_SLOPE * x;
}

// =====================================================================
// K1: h[n,p,0:256] = feat[n,:] @ Ws[p]   (f32 WMMA 16x16x4)
// block = 128 threads (4 waves). Block tile: 16 rows x 256 cols.
// wave w covers cols [64w, 64w+64) as 4 WMMA col-tiles. K = 128 (32 steps).
// =====================================================================
__global__ __launch_bounds__(128) void gat_gemm_hW(const float* __restrict__ feat,
                                                   const float* __restrict__ Ws,
                                                   float* __restrict__ h) {
  const int lane = threadIdx.x & 31;
  const int wave = threadIdx.x >> 5;
  const int row0 = blockIdx.x * 16;        // N/16 = 3125 exact
  const int p    = blockIdx.y;
  const int colb = wave * 64;
  const int m    = lane & 15;              // A row / B col within tile
  const int koff = (lane >> 4) << 1;       // lanes 16-31 hold K+2,K+3

  const float* A  = feat + (size_t)(row0 + m) * DIN + koff;
  const float* Bp = Ws + (size_t)p * DIN * HO + colb + m;

  v8f acc0 = {}, acc1 = {}, acc2 = {}, acc3 = {};
  for (int k = 0; k < DIN; k += 4) {
    v2f a = *(const v2f*)(A + k);
    const float* Bk = Bp + (size_t)(k + koff) * HO;
    v2f b0, b1, b2, b3;
    b0.x = Bk[0];   b0.y = Bk[HO];
    b1.x = Bk[16];  b1.y = Bk[HO + 16];
    b2.x = Bk[32];  b2.y = Bk[HO + 32];
    b3.x = Bk[48];  b3.y = Bk[HO + 48];
    acc0 = WMMA_F32(a, b0, acc0);
    acc1 = WMMA_F32(a, b1, acc1);
    acc2 = WMMA_F32(a, b2, acc2);
    acc3 = WMMA_F32(a, b3, acc3);
  }

  const int rg = (lane >> 4) * 8;          // C layout: VGPR r -> M = r (+8 for hi lanes)
#pragma unroll
  for (int r = 0; r < 8; ++r) {
    float* dst = h + (((size_t)(row0 + rg + r)) * PP + p) * HO + colb + m;
    dst[0]  = acc0[r];
    dst[16] = acc1[r];
    dst[32] = acc2[r];
    dst[48] = acc3[r];
  }
}

// =====================================================================
// K2: el/er[n,p,hd] = sum_o h[n,p,hd,o] * attn_{l,r}[p,hd,o]
// =====================================================================
__global__ void gat_elr(const float* __restrict__ h,
                        const float* __restrict__ al, const float* __restrict__ ar,
                        float* __restrict__ el, float* __restrict__ er) {
  int idx = blockIdx.x * blockDim.x + threadIdx.x;   // over N*P*H
  if (idx >= NN * PP * HH) return;
  int hd = idx % HH;
  int np = idx / HH;
  int p  = np % PP;
  const float* hr  = h  + (size_t)np * HO + hd * OO;
  const float* alp = al + (p * HH + hd) * OO;
  const float* arp = ar + (p * HH + hd) * OO;
  float sl = 0.f, sr = 0.f;
#pragma unroll
  for (int o = 0; o < OO; ++o) { float v = hr[o]; sl += v * alp[o]; sr += v * arp[o]; }
  el[idx] = sl;
  er[idx] = sr;
}

// =====================================================================
// K3: segment max over incoming edges (exact, via monotone u32 atomicMax)
// =====================================================================
__global__ void gat_emax(const long long* __restrict__ edges,
                         const float* __restrict__ el, const float* __restrict__ er,
                         unsigned* __restrict__ m_enc) {
  int e = blockIdx.x * blockDim.x + threadIdx.x;
  int p = blockIdx.y;
  if (e >= EE) return;
  const long long* ep = edges + (size_t)p * 2 * EE;
  int src = (int)ep[e];
  int dst = (int)ep[EE + e];
  const float* elr = el + ((size_t)src * PP + p) * HH;
  const float* err = er + ((size_t)dst * PP + p) * HH;
  unsigned* mr = m_enc + ((size_t)dst * PP + p) * HH;
#pragma unroll
  for (int hd = 0; hd < HH; ++hd) {
    float x = leaky(elr[hd] + err[hd]);
    atomicMax(&mr[hd], enc_f32(x));
  }
}

// =====================================================================
// K4: segment sum of exp(e - m[dst])  (recompute e -> no 38MB edge buffer)
// =====================================================================
__global__ void gat_esum(const long long* __restrict__ edges,
                         const float* __restrict__ el, const float* __restrict__ er,
                         const unsigned* __restrict__ m_enc, float* __restrict__ ssum) {
  int e = blockIdx.x * blockDim.x + threadIdx.x;
  int p = blockIdx.y;
  if (e >= EE) return;
  const long long* ep = edges + (size_t)p * 2 * EE;
  int src = (int)ep[e];
  int dst = (int)ep[EE + e];
  const float* elr = el + ((size_t)src * PP + p) * HH;
  const float* err = er + ((size_t)dst * PP + p) * HH;
  const unsigned* mr = m_enc + ((size_t)dst * PP + p) * HH;
  float* sr = ssum + ((size_t)dst * PP + p) * HH;
#pragma unroll
  for (int hd = 0; hd < HH; ++hd) {
    float x = leaky(elr[hd] + err[hd]);
    atomicAdd(&sr[hd], __expf(x - dec_f32(mr[hd])));
  }
}

// =====================================================================
// K5: agg[dst,p,hd,:] += alpha * h[src,p,hd,:]   (1 thread per edge*head)
// =====================================================================
__global__ void gat_aggr(const long long* __restrict__ edges,
                         const float* __restrict__ el, const float* __restrict__ er,
                         const unsigned* __restrict__ m_enc, const float* __restrict__ ssum,
                         const float* __restrict__ h, float* __restrict__ agg) {
  int e  = blockIdx.x * blockDim.x + threadIdx.x;
  int hd = blockIdx.y;
  int p  = blockIdx.z;
  if (e >= EE) return;
  const long long* ep = edges + (size_t)p * 2 * EE;
  int src = (int)ep[e];
  int dst = (int)ep[EE + e];
  size_t si = (size_t)src * PP + p;
  size_t di = (size_t)dst * PP + p;
  float x = leaky(el[si * HH + hd] + er[di * HH + hd]);
  float alpha = __expf(x - dec_f32(m_enc[di * HH + hd])) / ssum[di * HH + hd];
  const float4* hp = (const float4*)(h + si * HO + hd * OO);
  float* ap = agg + di * HO + hd * OO;
#pragma unroll
  for (int q = 0; q < 8; ++q) {
    float4 v = hp[q];
    atomicAdd(ap + q * 4 + 0, alpha * v.x);
    atomicAdd(ap + q * 4 + 1, alpha * v.y);
    atomicAdd(ap + q * 4 + 2, alpha * v.z);
    atomicAdd(ap + q * 4 + 3, alpha * v.w);
  }
}

// =====================================================================
// K6: z = elu(agg + bias)  (in place over agg)
// =====================================================================
__global__ void gat_elu(float* __restrict__ agg, const float* __restrict__ bias) {
  size_t idx = (size_t)blockIdx.x * blockDim.x + threadIdx.x;  // over N*P*HO
  if (idx >= (size_t)NN * PP * HO) return;
  int c = (int)(idx % HO);
  int p = (int)((idx / HO) % PP);
  float v = agg[idx] + bias[p * HO + c];
  agg[idx] = v > 0.f ? v : expm1f(v);
}

// =====================================================================
// K7: semantic projection with WMMA: t = z @ sem_W1 (16x128 per block,
// K=256), then per-row reduce sum_c tanh(t+b1)*w2 -> atomicAdd wsum[p].
// =====================================================================
__global__ __launch_bounds__(128) void sem_gemm(const float* __restrict__ z,
                                                const float* __restrict__ W1,
                                                const float* __restrict__ b1,
                                                const float* __restrict__ w2,
                                                float* __restrict__ wsum) {
  __shared__ float tile[16][HID + 4];
  const int lane = threadIdx.x & 31;
  const int wave = threadIdx.x >> 5;
  const int R0   = blockIdx.x * 16;        // (N*P)/16 = 9375 exact; row R = n*P+p
  const int colb = wave * 32;              // each wave: 2 col-tiles of 16
  const int m    = lane & 15;
  const int koff = (lane >> 4) << 1;

  const float* A  = z  + (size_t)(R0 + m) * HO + koff;
  const float* Bp = W1 + colb + m;

  v8f acc0 = {}, acc1 = {};
  for (int k = 0; k < HO; k += 4) {
    v2f a = *(const v2f*)(A + k);
    const float* Bk = Bp + (size_t)(k + koff) * HID;
    v2f b0, b1v;
    b0.x  = Bk[0];   b0.y  = Bk[HID];
    b1v.x = Bk[16];  b1v.y = Bk[HID + 16];
    acc0 = WMMA_F32(a, b0, acc0);
    acc1 = WMMA_F32(a, b1v, acc1);
  }

  const int rg = (lane >> 4) * 8;
#pragma unroll
  for (int r = 0; r < 8; ++r) {
    tile[rg + r][colb + m]      = acc0[r];
    tile[rg + r][colb + 16 + m] = acc1[r];
  }
  __syncthreads();

  if (threadIdx.x < 16) {
    int R = R0 + threadIdx.x;
    float s = 0.f;
    for (int c = 0; c < HID; ++c) s += tanhf(tile[threadIdx.x][c] + b1[c]) * w2[c];
    atomicAdd(&wsum[R % PP], s);
  }
}

// =====================================================================
// K8: beta = softmax(wsum/N); out[n,:] = sum_p beta[p] * z[n,p,:]
// =====================================================================
__global__ void han_out(const float* __restrict__ z, const float* __restrict__ wsum,
                        float* __restrict__ out) {
  size_t idx = (size_t)blockIdx.x * blockDim.x + threadIdx.x;  // over N*HO
  if (idx >= (size_t)NN * HO) return;
  float w0 = wsum[0] / (float)NN;
  float w1 = wsum[1] / (float)NN;
  float w2 = wsum[2] / (float)NN;
  float mx = fmaxf(w0, fmaxf(w1, w2));
  float e0 = __expf(w0 - mx), e1 = __expf(w1 - mx), e2 = __expf(w2 - mx);
  float inv = 1.f / (e0 + e1 + e2);
  size_t n = idx / HO;
  int c = (int)(idx % HO);
  const float* zr = z + (n * PP) * (size_t)HO + c;
  out[idx] = (e0 * zr[0] + e1 * zr[HO] + e2 * zr[2 * HO]) * inv;
}

// =====================================================================
extern "C" void kernel_launch(void* const* d_in, const int* in_sizes, int n_in,
                              void* d_out, int out_size, void* d_ws, size_t ws_size,
                              hipStream_t stream) {
  (void)in_sizes; (void)n_in; (void)out_size; (void)ws_size;

  const float*     feat  = (const float*)d_in[0];
  const long long* edges = (const long long*)d_in[1];
  const float*     Ws    = (const float*)d_in[2];
  const float*     al    = (const float*)d_in[3];
  const float*     ar    = (const float*)d_in[4];
  const float*     bias  = (const float*)d_in[5];
  const float*     W1    = (const float*)d_in[6];
  const float*     b1    = (const float*)d_in[7];
  const float*     w2    = (const float*)d_in[8];
  float* out = (float*)d_out;

  // -------- workspace layout (~326 MB) --------
  char* ws = (char*)d_ws;
  size_t off = 0;
  float*    h     = (float*)(ws + off); off += (size_t)NN * PP * HO * 4;   // 153.6 MB
  float*    agg   = (float*)(ws + off); off += (size_t)NN * PP * HO * 4;   // 153.6 MB (reused as z)
  float*    el    = (float*)(ws + off); off += (size_t)NN * PP * HH * 4;
  float*    er    = (float*)(ws + off); off += (size_t)NN * PP * HH * 4;
  unsigned* m_enc = (unsigned*)(ws + off); off += (size_t)NN * PP * HH * 4;
  float*    ssum  = (float*)(ws + off); off += (size_t)NN * PP * HH * 4;
  float*    wsum  = (float*)(ws + off); off += 16;

  // zero-init accumulators (m_enc==0 encodes the minimal float in enc order)
  hipMemsetAsync(agg,   0, (size_t)NN * PP * HO * 4, stream);
  hipMemsetAsync(m_enc, 0, (size_t)NN * PP * HH * 4, stream);
  hipMemsetAsync(ssum,  0, (size_t)NN * PP * HH * 4, stream);
  hipMemsetAsync(wsum,  0, 16, stream);

  // 1) h = feat @ Ws[p]  (WMMA f32)
  gat_gemm_hW<<<dim3(NN / 16, PP), 128, 0, stream>>>(feat, Ws, h);
  // 2) attention logits per node
  gat_elr<<<(NN * PP * HH + 255) / 256, 256, 0, stream>>>(h, al, ar, el, er);
  // 3) segment max
  gat_emax<<<dim3((EE + 255) / 256, PP), 256, 0, stream>>>(edges, el, er, m_enc);
  // 4) segment sum of exp
  gat_esum<<<dim3((EE + 255) / 256, PP), 256, 0, stream>>>(edges, el, er, m_enc, ssum);
  // 5) weighted aggregation (dominant phase; L2-resident gathers + atomics)
  gat_aggr<<<dim3((EE + 255) / 256, HH, PP), 256, 0, stream>>>(edges, el, er, m_enc, ssum, h, agg);
  // 6) z = elu(agg + bias) in place
  gat_elu<<<(NN * PP * HO) / 256, 256, 0, stream>>>(agg, bias);
  // 7) semantic scores (WMMA f32 + LDS reduce)
  sem_gemm<<<(NN * PP) / 16, 128, 0, stream>>>(agg, W1, b1, w2, wsum);
  // 8) softmax over paths + weighted combine
  han_out<<<(NN * HO) / 256, 256, 0, stream>>>(agg, wsum, out);
}